// SqueezeExcitationCUDA_86414741995572
// MI455X (gfx1250) — compile-verified
//
#include <hip/hip_runtime.h>
#include <hip/hip_bf16.h>

typedef __attribute__((ext_vector_type(2))) float v2f;
typedef __attribute__((ext_vector_type(4))) float f4;
typedef __attribute__((ext_vector_type(8))) float v8f;

#define B_DIM 32
#define C_DIM 256
#define R_DIM 64
#define HW    3136          // 56*56
#define HW4   784           // HW / 4
#define ROWS  (B_DIM * C_DIM)   // 8192

// ---------------------------------------------------------------------------
// Kernel 1: squeeze — global average pool over H,W.  One block per (b,c) row.
// x is [B, C, H, W] fp32; row = b*C + c; writes s[row] = mean.
// Default (RT) temporal hint: leaves x resident in the 192MB L2 for kernel 3.
// ---------------------------------------------------------------------------
__global__ __launch_bounds__(256)
void se_squeeze(const float* __restrict__ x, float* __restrict__ s) {
    const int row = blockIdx.x;
    const f4* p4 = (const f4*)(x + (size_t)row * HW);
    float acc = 0.0f;
    for (int i = threadIdx.x; i < HW4; i += 256) {
        f4 v = p4[i];
        acc += (v.x + v.y) + (v.z + v.w);
    }
    // wave32 shuffle reduction
    #pragma unroll
    for (int off = 16; off > 0; off >>= 1)
        acc += __shfl_xor(acc, off, 32);
    __shared__ float red[8];
    const int lane = threadIdx.x & 31;
    const int wave = threadIdx.x >> 5;
    if (lane == 0) red[wave] = acc;
    __syncthreads();
    if (threadIdx.x == 0) {
        float t = 0.0f;
        #pragma unroll
        for (int i = 0; i < 8; ++i) t += red[i];
        s[row] = t * (1.0f / (float)HW);
    }
}

// ---------------------------------------------------------------------------
// Kernel 2: tiny MLP on the pooled vector, via V_WMMA_F32_16X16X4_F32.
//   h[b,r] = swish( sum_c s[b,c]*w1[r,c] + b1[r] )      M=32, N=64,  K=256
//   e[b,c] = sigmoid( sum_r h[b,r]*w2[c,r] + b2[c] )    M=32, N=256, K=64
// Single workgroup, 8 waves; each wave owns 16x16 output tiles.
// A-frag (16x4 f32): lane 0-15 -> M rows, a[0]/a[1] = K0/K1 (lanes<16) or
// K2/K3 (lanes>=16).  B-frag mirrored with N on lanes.
// C/D layout: lane = N%16, vgpr v -> M = v + (lane>=16 ? 8 : 0).
// ---------------------------------------------------------------------------
__global__ __launch_bounds__(256)
void se_mlp(const float* __restrict__ s,
            const float* __restrict__ w1, const float* __restrict__ b1,
            const float* __restrict__ w2, const float* __restrict__ b2,
            float* __restrict__ e) {
    __shared__ float h_lds[B_DIM * R_DIM];   // 8 KB

    const int lane  = threadIdx.x & 31;
    const int wave  = threadIdx.x >> 5;
    const int lrow  = lane & 15;             // tile-local M (for A) / N (for B,D)
    const int khalf = (lane >> 4) * 2;       // 0 for lanes 0-15, 2 for 16-31
    const int mhalf = (lane >> 4) * 8;       // D-matrix M offset per lane half

    // ---- GEMM1 + swish -> h in LDS.  wave -> (mt = wave>>2, nt = wave&3)
    {
        const int mt = wave >> 2;            // 0..1   (batch tiles)
        const int nt = wave & 3;             // 0..3   (R tiles)
        const int m  = mt * 16 + lrow;       // batch row for A-frag
        const int n  = nt * 16 + lrow;       // r index for B-frag
        v8f acc = {};
        for (int kk = 0; kk < C_DIM; kk += 4) {
            const int k0 = kk + khalf;
            v2f a, b;
            a.x = s[m * C_DIM + k0];
            a.y = s[m * C_DIM + k0 + 1];
            b.x = w1[n * C_DIM + k0];        // B[k][n] = w1[n][k]
            b.y = w1[n * C_DIM + k0 + 1];
            acc = __builtin_amdgcn_wmma_f32_16x16x4_f32(
                false, a, false, b, (short)0, acc, false, false);
        }
        const float bias = b1[n];
        const int mbase = mt * 16 + mhalf;
        #pragma unroll
        for (int v = 0; v < 8; ++v) {
            float val = acc[v] + bias;
            h_lds[(mbase + v) * R_DIM + n] = val / (1.0f + __expf(-val));
        }
    }
    __syncthreads();

    // ---- GEMM2 + sigmoid -> e[32][256] in global scratch.
    {
        const int mt = wave >> 2;
        const int m  = mt * 16 + lrow;
        const int mbase = mt * 16 + mhalf;
        for (int t = 0; t < 4; ++t) {
            const int nt = (wave & 3) + t * 4;   // 0..15 (C tiles)
            const int n  = nt * 16 + lrow;       // channel 0..255
            v8f acc = {};
            for (int kk = 0; kk < R_DIM; kk += 4) {
                const int k0 = kk + khalf;
                v2f a, b;
                a.x = h_lds[m * R_DIM + k0];
                a.y = h_lds[m * R_DIM + k0 + 1];
                b.x = w2[n * R_DIM + k0];        // B[k][n] = w2[n][k]
                b.y = w2[n * R_DIM + k0 + 1];
                acc = __builtin_amdgcn_wmma_f32_16x16x4_f32(
                    false, a, false, b, (short)0, acc, false, false);
            }
            const float bias = b2[n];
            #pragma unroll
            for (int v = 0; v < 8; ++v) {
                float val = acc[v] + bias;
                e[(mbase + v) * C_DIM + n] = 1.0f / (1.0f + __expf(-val));
            }
        }
    }
}

// ---------------------------------------------------------------------------
// Kernel 3: excite — out[b,c,:] = x[b,c,:] * e[b,c].
// x loads hit L2 (left resident by kernel 1); stores are non-temporal so the
// output stream doesn't evict x from L2 mid-kernel.
// ---------------------------------------------------------------------------
__global__ __launch_bounds__(256)
void se_excite(const float* __restrict__ x, const float* __restrict__ e,
               float* __restrict__ out) {
    const int row = blockIdx.x;
    const float scale = e[row];
    const f4* xin = (const f4*)(x + (size_t)row * HW);
    f4* o = (f4*)(out + (size_t)row * HW);
    for (int i = threadIdx.x; i < HW4; i += 256) {
        f4 v = xin[i];
        v.x *= scale; v.y *= scale; v.z *= scale; v.w *= scale;
        __builtin_nontemporal_store(v, o + i);
    }
}

// ---------------------------------------------------------------------------
extern "C" void kernel_launch(void* const* d_in, const int* in_sizes, int n_in,
                              void* d_out, int out_size, void* d_ws, size_t ws_size,
                              hipStream_t stream) {
    const float* x  = (const float*)d_in[0];   // [32,256,56,56]
    const float* w1 = (const float*)d_in[1];   // [64,256]
    const float* b1 = (const float*)d_in[2];   // [64]
    const float* w2 = (const float*)d_in[3];   // [256,64]
    const float* b2 = (const float*)d_in[4];   // [256]
    float* out = (float*)d_out;

    float* s = (float*)d_ws;                   // [32*256]
    float* e = s + ROWS;                       // [32*256]

    se_squeeze<<<ROWS, 256, 0, stream>>>(x, s);
    se_mlp<<<1, 256, 0, stream>>>(s, w1, b1, w2, b2, e);
    se_excite<<<ROWS, 256, 0, stream>>>(x, e, out);
}